// CubicKANLayerBlock_962072674707
// MI455X (gfx1250) — compile-verified
//
#include <hip/hip_runtime.h>
#include <hip/hip_bf16.h>

typedef __attribute__((ext_vector_type(2))) float v2f;
typedef __attribute__((ext_vector_type(8))) float v8f;

#define B_SZ   1024
#define D_IN   256
#define D_OUT  512
#define NKNOT  64

// Spline domain constants (computed in double as the Python reference does)
#define PHI_MIN_F   (-0.1f)
#define PHI_INV_D   ((float)(63.0 / ((1.1 + 0.02 * 511.0) - (-0.1))))   // 63/11.42
#define BIG_MIN_F   (-5.0f)
#define BIG_INV_D   ((float)(63.0 / 10.0))                               // 6.3

// ---------------------------------------------------------------------------
// Kernel 1: residual GEMM  R[b,o] = sum_k x_original[b,k] * W[k,o]
// FP32 WMMA: V_WMMA_F32_16X16X4_F32. One wave computes a 16x64 output tile
// (4 N-subtiles sharing one A fragment per K step).
// Grid: 64 blocks x 256 threads (8 waves/block), 512 wave-jobs total.
// ---------------------------------------------------------------------------
__global__ __launch_bounds__(256) void kan_residual_gemm(
    const float* __restrict__ X,   // [1024, 256] row-major
    const float* __restrict__ W,   // [256, 512]  row-major
    float* __restrict__ Out)       // [1024, 512]
{
    const int lane = threadIdx.x & 31;
    const int job  = blockIdx.x * 8 + (threadIdx.x >> 5);   // 0..511
    const int tm   = job >> 3;            // 0..63  (16-row tile)
    const int tn4  = job & 7;             // 0..7   (64-col supertile)
    const int row0 = tm << 4;
    const int col0 = tn4 << 6;

    const int m  = lane & 15;             // A row within tile / B column within subtile
    const int kk = (lane >> 4) << 1;      // lanes 0-15 -> K {0,1}; lanes 16-31 -> K {2,3}

    v8f acc0 = {}, acc1 = {}, acc2 = {}, acc3 = {};

    const float* Arow = X + (size_t)(row0 + m) * D_IN + kk;

    for (int k = 0; k < D_IN; k += 4) {
        // A fragment: 16x4 f32, lane holds A[m][k+kk], A[m][k+kk+1]
        v2f a = *(const v2f*)(Arow + k);

        const float* wr0 = W + (size_t)(k + kk) * D_OUT + col0 + m;
        const float* wr1 = wr0 + D_OUT;

        v2f b0 = { wr0[0],  wr1[0]  };
        v2f b1 = { wr0[16], wr1[16] };
        v2f b2 = { wr0[32], wr1[32] };
        v2f b3 = { wr0[48], wr1[48] };

        acc0 = __builtin_amdgcn_wmma_f32_16x16x4_f32(false, a, false, b0, (short)0, acc0, false, false);
        acc1 = __builtin_amdgcn_wmma_f32_16x16x4_f32(false, a, false, b1, (short)0, acc1, false, false);
        acc2 = __builtin_amdgcn_wmma_f32_16x16x4_f32(false, a, false, b2, (short)0, acc2, false, false);
        acc3 = __builtin_amdgcn_wmma_f32_16x16x4_f32(false, a, false, b3, (short)0, acc3, false, false);
    }

    // D layout: VGPR r -> M = r + 8*(lane>=16), N = lane&15
    const int nn = lane & 15;
    const int mh = (lane >> 4) << 3;
#pragma unroll
    for (int r = 0; r < 8; ++r) {
        float* p = Out + (size_t)(row0 + mh + r) * D_OUT + col0 + nn;
        p[0]  = acc0[r];
        p[16] = acc1[r];
        p[32] = acc2[r];
        p[48] = acc3[r];
    }
}

// ---------------------------------------------------------------------------
// Spline machinery: per-interval cubic coefficients in t (local param),
// plus linear-extrapolation entries at tab[0] (left) and tab[64] (right).
// y(traw) = c0 + c1*t + c2*t^2 + c3*t^3 with t = traw - clamp(floor(traw),0,63)
// and table index = clamp(floor(traw)+1, 0, 64).
// ---------------------------------------------------------------------------
__device__ __forceinline__ float4 make_coeffs(const float* __restrict__ v, int k)
{
    if (k == 0) {                                   // left linear extrapolation
        return make_float4(v[0], v[1] - v[0], 0.0f, 0.0f);
    }
    if (k == NKNOT) {                               // right linear extrapolation
        return make_float4(v[NKNOT - 1], v[NKNOT - 1] - v[NKNOT - 2], 0.0f, 0.0f);
    }
    const int j = k - 1;                            // interval j in [0,62]
    const float v0 = v[j];
    const float v1 = v[j + 1];
    const int im1 = (j > 0) ? (j - 1) : 0;
    const int ip2 = (j + 2 < NKNOT) ? (j + 2) : (NKNOT - 1);
    const float m0h = 0.5f * (v1 - v[im1]);         // m0 * h  (h = delta)
    const float m1h = 0.5f * (v[ip2] - v0);         // m1 * h
    const float d = v1 - v0;
    return make_float4(v0,
                       m0h,
                       3.0f * d - 2.0f * m0h - m1h,
                       -2.0f * d + m0h + m1h);
}

__device__ __forceinline__ float spline_tab(const float4* __restrict__ tab, float traw)
{
    const int j = (int)floorf(traw);
    int tb = j + 1;
    tb = tb < 0 ? 0 : (tb > NKNOT ? NKNOT : tb);
    int bs = j < 0 ? 0 : (j > NKNOT - 1 ? NKNOT - 1 : j);
    const float t = traw - (float)bs;
    const float4 c = tab[tb];                       // ds_load_b128 gather
    return fmaf(fmaf(fmaf(c.w, t, c.z), t, c.y), t, c.x);
}

// ---------------------------------------------------------------------------
// Kernel 2: one block per batch row b.
//   s[b,o]   = sum_i lambda[i] * phi(x[b,i] + eta*o)
//   out[b,o] = Phi(s[b,o]) + residual (already in Out from kernel 1)
// Thread t handles o = t and o = t + 256.
// ---------------------------------------------------------------------------
__global__ __launch_bounds__(256) void kan_spline_main(
    const float* __restrict__ X,        // [1024, 256]
    const float* __restrict__ phi_v,    // [64]
    const float* __restrict__ Phi_v,    // [64]
    const float* __restrict__ lam,      // [256]
    const float* __restrict__ eta_p,    // [1]
    float* __restrict__ Out)            // [1024, 512] (contains residual)
{
    __shared__ float4 sphi[NKNOT + 1];
    __shared__ float4 sPhi[NKNOT + 1];
    __shared__ float  xs[D_IN];
    __shared__ float  ls[D_IN];

    const int tid = threadIdx.x;
    const int b   = blockIdx.x;

    if (tid <= NKNOT) {
        sphi[tid] = make_coeffs(phi_v, tid);
    } else if (tid <= 2 * NKNOT + 1) {
        sPhi[tid - (NKNOT + 1)] = make_coeffs(Phi_v, tid - (NKNOT + 1));
    }
    xs[tid] = X[(size_t)b * D_IN + tid];
    ls[tid] = lam[tid];
    __syncthreads();

    const float eta = eta_p[0];
    const int   o0  = tid;
    const int   o1  = tid + 256;
    const float sh0 = eta * (float)o0;              // eta * q
    const float sh1 = eta * (float)o1;
    const float off = -PHI_MIN_F * PHI_INV_D;       // maps shifted -> traw

    float acc0 = 0.0f, acc1 = 0.0f;

#pragma unroll 4
    for (int i = 0; i < D_IN; ++i) {
        const float xi = xs[i];
        const float li = ls[i];
        const float t0 = fmaf(xi + sh0, PHI_INV_D, off);
        const float t1 = fmaf(xi + sh1, PHI_INV_D, off);
        acc0 = fmaf(li, spline_tab(sphi, t0), acc0);
        acc1 = fmaf(li, spline_tab(sphi, t1), acc1);
    }

    const float offB = -BIG_MIN_F * BIG_INV_D;
    const float y0 = spline_tab(sPhi, fmaf(acc0, BIG_INV_D, offB));
    const float y1 = spline_tab(sPhi, fmaf(acc1, BIG_INV_D, offB));

    float* orow = Out + (size_t)b * D_OUT;
    orow[o0] = y0 + orow[o0];
    orow[o1] = y1 + orow[o1];
}

// ---------------------------------------------------------------------------
// Launch
// ---------------------------------------------------------------------------
extern "C" void kernel_launch(void* const* d_in, const int* in_sizes, int n_in,
                              void* d_out, int out_size, void* d_ws, size_t ws_size,
                              hipStream_t stream)
{
    const float* x       = (const float*)d_in[0];
    const float* x_orig  = (const float*)d_in[1];
    const float* phi_v   = (const float*)d_in[2];
    const float* Phi_v   = (const float*)d_in[3];
    const float* lambdas = (const float*)d_in[4];
    const float* eta     = (const float*)d_in[5];
    const float* W       = (const float*)d_in[6];
    float* out = (float*)d_out;

    // Residual GEMM via FP32 WMMA writes Out first; spline kernel adds on top.
    kan_residual_gemm<<<64, 256, 0, stream>>>(x_orig, W, out);
    kan_spline_main<<<B_SZ, 256, 0, stream>>>(x, phi_v, Phi_v, lambdas, eta, out);
}